// GAT_30253749633692
// MI455X (gfx1250) — compile-verified
//
#include <hip/hip_runtime.h>
#include <hip/hip_bf16.h>

typedef float v2f __attribute__((ext_vector_type(2)));
typedef float v8f __attribute__((ext_vector_type(8)));

#define NNODES 50000
#define NEDGES 800000
#define NGRAPH 64
#define HID 128
#define NCLS 10

// ---------------------------------------------------------------------------
// fp32 WMMA GEMM: C[n x 128] = A[n x 128] * B[128 x 128] + bias
// One wave per 16x16 output tile, 8 waves/block cover a 16-row stripe.
// Uses V_WMMA_F32_16X16X4_F32 (fp32 in, fp32 acc) - GEMM is not the
// bottleneck here, so keep full reference precision.
// ---------------------------------------------------------------------------
__global__ void __launch_bounds__(256) gemm128_wmma(
    const float* __restrict__ A, const float* __restrict__ B,
    const float* __restrict__ bias, float* __restrict__ C, int nrows) {
  __shared__ float As[16 * 132];  // 132-float pitch -> no LDS bank conflicts
  int row0 = blockIdx.x << 4;
  // cooperative load of 16x128 A tile
  for (int i = threadIdx.x; i < 16 * 32; i += 256) {
    int r = i >> 5, c4 = i & 31;
    float4 v = ((const float4*)(A + (size_t)(row0 + r) * HID))[c4];
    *((float4*)(As + r * 132) + c4) = v;
  }
  __syncthreads();
  int wave = threadIdx.x >> 5;   // output column tile 0..7
  int lane = threadIdx.x & 31;
  int half = lane >> 4;          // selects K pair (0,1) vs (2,3)
  int l16  = lane & 15;
  int col  = (wave << 4) + l16;
  v8f acc = {};
  const float* Ar = As + l16 * 132 + 2 * half;
  const float* Bp = B + (size_t)(2 * half) * HID + col;
#pragma unroll
  for (int k = 0; k < HID; k += 4) {
    v2f a, b;
    a.x = Ar[k];
    a.y = Ar[k + 1];
    b.x = Bp[(size_t)k * HID];
    b.y = Bp[(size_t)(k + 1) * HID];
    acc = __builtin_amdgcn_wmma_f32_16x16x4_f32(false, a, false, b,
                                                (short)0, acc, false, false);
  }
  float bv = bias[col];
#pragma unroll
  for (int r = 0; r < 8; ++r) {
    int row = row0 + r + 8 * half;  // C frag: vgpr r -> M=r (+8 for hi half)
    if (row < nrows) C[(size_t)row * HID + col] = acc[r] + bv;
  }
}

// ---- monotone float <-> uint mapping so atomicMax(u32) == float max --------
__device__ __forceinline__ unsigned ordEnc(float f) {
  unsigned u = __float_as_uint(f);
  return (u & 0x80000000u) ? ~u : (u | 0x80000000u);
}
__device__ __forceinline__ float ordDec(unsigned u) {
  return (u & 0x80000000u) ? __uint_as_float(u & 0x7fffffffu)
                           : __uint_as_float(~u);
}

// ---------------------------------------------------------------------------
// Edge scores: e[k,h] = sum_c att[h,c]*leakyrelu(xl[src,h,c]+xr[dst,h,c])
// One wave per edge; lane handles 4 channels (float4 of the 512B row).
// Fused segment-max via ordered-uint atomicMax.
// ---------------------------------------------------------------------------
__global__ void __launch_bounds__(256) edge_score(
    const float* __restrict__ xl, const float* __restrict__ xr,
    const int* __restrict__ src, const int* __restrict__ dst,
    const float* __restrict__ att, float* __restrict__ e,
    unsigned* __restrict__ mOrd, int nE, int nTot) {
  int wid = blockIdx.x * (blockDim.x >> 5) + (threadIdx.x >> 5);
  if (wid >= nTot) return;
  int lane = threadIdx.x & 31;
  int s, d;
  if (wid < nE) { s = src[wid]; d = dst[wid]; } else { s = wid - nE; d = s; }
  float4 a = ((const float4*)(xl + (size_t)s * HID))[lane];
  float4 b = ((const float4*)(xr + (size_t)d * HID))[lane];
  float4 t = ((const float4*)att)[lane];
  float s0 = a.x + b.x; s0 = s0 > 0.f ? s0 : 0.2f * s0;
  float s1 = a.y + b.y; s1 = s1 > 0.f ? s1 : 0.2f * s1;
  float s2 = a.z + b.z; s2 = s2 > 0.f ? s2 : 0.2f * s2;
  float s3 = a.w + b.w; s3 = s3 > 0.f ? s3 : 0.2f * s3;
  float v = s0 * t.x + s1 * t.y + s2 * t.z + s3 * t.w;
  v += __shfl_down(v, 4, 8);   // reduce over the 8 lanes of one head
  v += __shfl_down(v, 2, 8);
  v += __shfl_down(v, 1, 8);
  if ((lane & 7) == 0) {
    int h = lane >> 3;
    e[(size_t)wid * 4 + h] = v;
    atomicMax(&mOrd[(size_t)d * 4 + h], ordEnc(v));
  }
}

// exp(e - max[dst]) and segment-sum of denominators
__global__ void __launch_bounds__(256) edge_exp(
    const int* __restrict__ dst, float* __restrict__ e,
    const unsigned* __restrict__ mOrd, float* __restrict__ denom,
    int nE, int nTot) {
  long long i = (long long)blockIdx.x * blockDim.x + threadIdx.x;
  if (i >= (long long)nTot * 4) return;
  int edge = (int)(i >> 2), h = (int)(i & 3);
  int d = (edge < nE) ? dst[edge] : (edge - nE);
  float m  = ordDec(mOrd[(size_t)d * 4 + h]);
  float ee = expf(e[i] - m);
  e[i] = ee;
  atomicAdd(&denom[(size_t)d * 4 + h], ee);
}

// out[dst] += alpha * xl[src]; one wave per edge, 4 f32 atomics per lane
__global__ void __launch_bounds__(256) edge_aggregate(
    const float* __restrict__ xl, const float* __restrict__ e,
    const float* __restrict__ denom, const int* __restrict__ src,
    const int* __restrict__ dst, float* __restrict__ acc, int nE, int nTot) {
  int wid = blockIdx.x * (blockDim.x >> 5) + (threadIdx.x >> 5);
  if (wid >= nTot) return;
  int lane = threadIdx.x & 31;
  int s, d;
  if (wid < nE) { s = src[wid]; d = dst[wid]; } else { s = wid - nE; d = s; }
  int h = lane >> 3;
  float alpha = e[(size_t)wid * 4 + h] / denom[(size_t)d * 4 + h];
  float4 a = ((const float4*)(xl + (size_t)s * HID))[lane];
  float* o = acc + (size_t)d * HID + lane * 4;
  atomicAdd(o + 0, a.x * alpha);
  atomicAdd(o + 1, a.y * alpha);
  atomicAdd(o + 2, a.z * alpha);
  atomicAdd(o + 3, a.w * alpha);
}

// h = elu(h + bias), in place
__global__ void __launch_bounds__(256) finalize_elu(
    float* __restrict__ h, const float* __restrict__ bias, int n) {
  long long i = (long long)blockIdx.x * blockDim.x + threadIdx.x;
  if (i >= (long long)n * HID) return;
  float v = h[i] + bias[(int)(i & (HID - 1))];
  h[i] = v > 0.f ? v : (expf(v) - 1.f);
}

// global mean-pool accumulation (sums + counts)
__global__ void __launch_bounds__(256) pool_kernel(
    const float* __restrict__ h, const int* __restrict__ batch,
    float* __restrict__ sums, float* __restrict__ cnt, int n) {
  int wid = blockIdx.x * (blockDim.x >> 5) + (threadIdx.x >> 5);
  if (wid >= n) return;
  int lane = threadIdx.x & 31;
  int g = batch[wid];
  float4 v = ((const float4*)(h + (size_t)wid * HID))[lane];
  float* o = sums + (size_t)g * HID + lane * 4;
  atomicAdd(o + 0, v.x);
  atomicAdd(o + 1, v.y);
  atomicAdd(o + 2, v.z);
  atomicAdd(o + 3, v.w);
  if (lane == 0) atomicAdd(&cnt[g], 1.0f);
}

// pooled @ lin_w + lin_b -> elu -> log_softmax ; one thread per graph
__global__ void head_kernel(const float* __restrict__ sums,
                            const float* __restrict__ cnt,
                            const float* __restrict__ lin_w,
                            const float* __restrict__ lin_b,
                            float* __restrict__ out) {
  int g = threadIdx.x;
  if (g >= NGRAPH) return;
  float c = cnt[g]; c = c > 1.f ? c : 1.f;
  float inv = 1.f / c;
  float l[NCLS];
#pragma unroll
  for (int j = 0; j < NCLS; ++j) l[j] = lin_b[j];
  for (int k = 0; k < HID; ++k) {
    float p = sums[g * HID + k] * inv;
#pragma unroll
    for (int j = 0; j < NCLS; ++j) l[j] += p * lin_w[k * NCLS + j];
  }
  float mx = -1e30f;
#pragma unroll
  for (int j = 0; j < NCLS; ++j) {
    l[j] = l[j] > 0.f ? l[j] : (expf(l[j]) - 1.f);
    mx = l[j] > mx ? l[j] : mx;
  }
  float se = 0.f;
#pragma unroll
  for (int j = 0; j < NCLS; ++j) se += expf(l[j] - mx);
  float lse = mx + logf(se);
#pragma unroll
  for (int j = 0; j < NCLS; ++j) out[g * NCLS + j] = l[j] - lse;
}

extern "C" void kernel_launch(void* const* d_in, const int* in_sizes, int n_in,
                              void* d_out, int out_size, void* d_ws, size_t ws_size,
                              hipStream_t stream) {
  (void)in_sizes; (void)n_in; (void)out_size; (void)ws_size;
  const float* x     = (const float*)d_in[0];
  const int*   ei    = (const int*)d_in[1];
  const int*   batch = (const int*)d_in[2];
  const float* Wl    = (const float*)d_in[3];
  const float* Wr    = (const float*)d_in[4];
  const float* bl    = (const float*)d_in[5];
  const float* br    = (const float*)d_in[6];
  const float* att   = (const float*)d_in[7];
  const float* bias  = (const float*)d_in[8];
  const float* lin_w = (const float*)d_in[9];
  const float* lin_b = (const float*)d_in[10];
  float* out = (float*)d_out;

  const int N = NNODES, E = NEDGES, ET = NEDGES + NNODES;
  const int* src = ei;
  const int* dst = ei + E;

  // workspace layout (~116 MB)
  char* w = (char*)d_ws;
  size_t off = 0;
  auto take = [&](size_t bytes) -> char* {
    char* p = w + off;
    off = (off + bytes + 255) & ~(size_t)255;
    return p;
  };
  float*    hA    = (float*)take((size_t)N * HID * 4);
  float*    hB    = (float*)take((size_t)N * HID * 4);
  float*    xl    = (float*)take((size_t)N * HID * 4);
  float*    xr    = (float*)take((size_t)N * HID * 4);
  float*    e     = (float*)take((size_t)ET * 4 * 4);
  unsigned* mOrd  = (unsigned*)take((size_t)N * 4 * 4);
  float*    denom = (float*)take((size_t)N * 4 * 4);
  float*    sums  = (float*)take((size_t)NGRAPH * HID * 4);
  float*    cnt   = (float*)take((size_t)NGRAPH * 4);

  dim3 blk(256);
  int gemmGrid   = N / 16;                                  // 3125
  int edgeBlocks = (ET + 7) / 8;                            // 8 waves/block
  int expBlocks  = (int)(((long long)ET * 4 + 255) / 256);
  int elemBlocks = (int)(((long long)N * HID + 255) / 256);

  const float* hcur = x;
  float* hnext = hA;
  for (int l = 0; l < 3; ++l) {
    gemm128_wmma<<<gemmGrid, blk, 0, stream>>>(hcur, Wl + (size_t)l * HID * HID,
                                               bl + l * HID, xl, N);
    gemm128_wmma<<<gemmGrid, blk, 0, stream>>>(hcur, Wr + (size_t)l * HID * HID,
                                               br + l * HID, xr, N);
    hipMemsetAsync(mOrd, 0, (size_t)N * 4 * 4, stream);
    hipMemsetAsync(denom, 0, (size_t)N * 4 * 4, stream);
    hipMemsetAsync(hnext, 0, (size_t)N * HID * 4, stream);
    edge_score<<<edgeBlocks, blk, 0, stream>>>(xl, xr, src, dst, att + l * HID,
                                               e, mOrd, E, ET);
    edge_exp<<<expBlocks, blk, 0, stream>>>(dst, e, mOrd, denom, E, ET);
    edge_aggregate<<<edgeBlocks, blk, 0, stream>>>(xl, e, denom, src, dst,
                                                   hnext, E, ET);
    finalize_elu<<<elemBlocks, blk, 0, stream>>>(hnext, bias + l * HID, N);
    hcur  = hnext;
    hnext = (hcur == hA) ? hB : hA;
  }
  hipMemsetAsync(sums, 0, (size_t)NGRAPH * HID * 4, stream);
  hipMemsetAsync(cnt, 0, (size_t)NGRAPH * 4, stream);
  pool_kernel<<<(N + 7) / 8, blk, 0, stream>>>(hcur, batch, sums, cnt, N);
  head_kernel<<<1, 64, 0, stream>>>(sums, cnt, lin_w, lin_b, out);
}